// HiddenLayer_23046794510859
// MI455X (gfx1250) — compile-verified
//
#include <hip/hip_runtime.h>

typedef __attribute__((ext_vector_type(2))) float v2f;
typedef __attribute__((ext_vector_type(8))) float v8f;
typedef int v4i __attribute__((vector_size(16)));   // matches builtin param type

constexpr int H  = 4096;   // hidden size
constexpr int NB = 8192;   // batch
constexpr int BM = 128;    // block tile M (batch rows)
constexpr int BN = 128;    // block tile N (output neurons)
constexpr int BK = 32;     // block tile K
constexpr int XS = 36;     // LDS row stride in floats (16B-aligned rows, conflict-free)
constexpr int KT = H / BK; // 128 K-tiles

// ---- CDNA5 async global->LDS path (guarded; sync fallback keeps compile OK) ----
#if defined(__has_builtin)
#if __has_builtin(__builtin_amdgcn_global_load_async_to_lds_b128)
#define HAVE_ASYNC_LDS 1
#endif
#endif

#ifdef HAVE_ASYNC_LDS
#define ASYNC_LDS_B128(gptr, lptr)                                             \
  __builtin_amdgcn_global_load_async_to_lds_b128(                              \
      (__attribute__((address_space(1))) v4i*)(gptr),                          \
      (__attribute__((address_space(3))) v4i*)(lptr), 0, 0)
#if __has_builtin(__builtin_amdgcn_s_wait_asynccnt)
#define WAIT_ASYNC(n) __builtin_amdgcn_s_wait_asynccnt(n)
#else
#define WAIT_ASYNC(n) asm volatile("s_wait_asynccnt %0" ::"i"(n) : "memory")
#endif
#endif

// ---------------------------------------------------------------------------
// Phase 1: W' = dale-clip(weight * sparse_mask).  dale_mask is a broadcast of
// its row 0 across all rows, so we only read dale_mask[0:H] (per column).
// ---------------------------------------------------------------------------
__global__ __launch_bounds__(256) void mask_weights_kernel(
    const float* __restrict__ weight,
    const int*   __restrict__ sparse,
    const int*   __restrict__ dale_row0,
    float*       __restrict__ wout) {
  const int idx = (blockIdx.x * 256 + threadIdx.x) * 4;   // H*H divisible by 1024
  const int j   = idx & (H - 1);                          // column index (4 consecutive)

  const float4 w = *(const float4*)(weight + idx);
  const int4   s = *(const int4*)(sparse + idx);
  const int4   d = *(const int4*)(dale_row0 + j);

  float4 o;
  o.x = s.x ? (d.x > 0 ? fmaxf(w.x, 0.f) : fminf(w.x, 0.f)) : 0.f;
  o.y = s.y ? (d.y > 0 ? fmaxf(w.y, 0.f) : fminf(w.y, 0.f)) : 0.f;
  o.z = s.z ? (d.z > 0 ? fmaxf(w.z, 0.f) : fminf(w.z, 0.f)) : 0.f;
  o.w = s.w ? (d.w > 0 ? fmaxf(w.w, 0.f) : fminf(w.w, 0.f)) : 0.f;
  *(float4*)(wout + idx) = o;
}

// ---------------------------------------------------------------------------
// Phase 2: out[b][o] = sum_k x[b][k] * W'[o][k] + bias[o]
// 128x128 block tile, 8 waves -> 4(M) x 2(N) wave grid, wave tile 32x64
// (2x4 grid of 16x16 fp32 WMMA accumulators), double-buffered LDS with
// async global->LDS copies.
// ---------------------------------------------------------------------------
__global__ __launch_bounds__(256, 2) void dale_gemm_kernel(
    const float* __restrict__ x,
    const float* __restrict__ wp,
    const float* __restrict__ bias,
    float*       __restrict__ out) {
  __shared__ float xlds[2][BM * XS];
  __shared__ float wlds[2][BN * XS];

  const int tid  = threadIdx.x;
  const int wave = tid >> 5;
  const int lane = tid & 31;
  const int lo   = lane & 15;     // M/N index within 16
  const int hi   = lane >> 4;     // K half-select for A/B frags
  const int wm   = wave & 3;      // wave M position: rows wm*32 .. wm*32+31
  const int wn   = wave >> 2;     // wave N position: cols wn*64 .. wn*64+63
  const int bm   = blockIdx.x * BM;
  const int bn   = blockIdx.y * BN;

  // global->LDS staging mapping: 8 threads per row of 32 floats, 32 rows/pass
  const int lr = tid >> 3;
  const int lc = (tid & 7) * 4;

  // accumulators initialized with bias (C/D layout: N = lane&15, same for all 8 VGPRs)
  v8f acc[2][4];
#pragma unroll
  for (int nt = 0; nt < 4; ++nt) {
    const float bv = bias[bn + wn * 64 + nt * 16 + lo];
#pragma unroll
    for (int mt = 0; mt < 2; ++mt)
#pragma unroll
      for (int v = 0; v < 8; ++v)
        acc[mt][nt][v] = bv;
  }

#ifdef HAVE_ASYNC_LDS
  // issue the 8 async b128 loads (4 x-tile + 4 w-tile rows) for K-tile kt
  auto issue_tile = [&](int kt, int bufi) {
    const int k0g = kt * BK;
#pragma unroll
    for (int p = 0; p < 4; ++p) {
      const int r = p * 32 + lr;
      ASYNC_LDS_B128(x  + (size_t)(bm + r) * H + k0g + lc, &xlds[bufi][r * XS + lc]);
      ASYNC_LDS_B128(wp + (size_t)(bn + r) * H + k0g + lc, &wlds[bufi][r * XS + lc]);
    }
  };
#else
  auto issue_tile = [&](int kt, int bufi) {
    const int k0g = kt * BK;
#pragma unroll
    for (int p = 0; p < 4; ++p) {
      const int r = p * 32 + lr;
      const float4 xv = *(const float4*)(x  + (size_t)(bm + r) * H + k0g + lc);
      const float4 wv = *(const float4*)(wp + (size_t)(bn + r) * H + k0g + lc);
      *(float4*)(&xlds[bufi][r * XS + lc]) = xv;
      *(float4*)(&wlds[bufi][r * XS + lc]) = wv;
    }
  };
#endif

  // prologue: prefetch K-tile 0 into buffer 0
  issue_tile(0, 0);

  for (int kt = 0; kt < KT; ++kt) {
    const int cur = kt & 1;

    // prefetch next tile into the other buffer (safe: barrier at end of kt-1
    // guaranteed all waves finished reading buffer (kt-1)&1 == (kt+1)&1)
    if (kt + 1 < KT) {
      issue_tile(kt + 1, cur ^ 1);
#ifdef HAVE_ASYNC_LDS
      WAIT_ASYNC(8);   // 8 newest = tile kt+1; everything older (tile kt) done
#endif
    } else {
#ifdef HAVE_ASYNC_LDS
      WAIT_ASYNC(0);   // drain: last tile's loads complete
#endif
    }
    __syncthreads();   // all threads' portions of buffer `cur` are in LDS

#pragma unroll
    for (int ks = 0; ks < BK / 4; ++ks) {
      const int k0 = ks * 4 + 2 * hi;   // lanes 0-15: K={k,k+1}; lanes 16-31: K={k+2,k+3}
      v2f a[2], b[4];
#pragma unroll
      for (int mt = 0; mt < 2; ++mt)
        a[mt] = *(const v2f*)(&xlds[cur][(wm * 32 + mt * 16 + lo) * XS + k0]);
#pragma unroll
      for (int nt = 0; nt < 4; ++nt)
        b[nt] = *(const v2f*)(&wlds[cur][(wn * 64 + nt * 16 + lo) * XS + k0]);
#pragma unroll
      for (int mt = 0; mt < 2; ++mt)
#pragma unroll
        for (int nt = 0; nt < 4; ++nt)
          acc[mt][nt] = __builtin_amdgcn_wmma_f32_16x16x4_f32(
              /*neg_a=*/false, a[mt], /*neg_b=*/false, b[nt],
              /*c_mod=*/(short)0, acc[mt][nt],
              /*reuse_a=*/false, /*reuse_b=*/false);
    }
    __syncthreads();   // all waves done reading buffer `cur`; next iter may refill it
  }

  // epilogue: C/D layout -> M = v + 8*hi, N = lo
#pragma unroll
  for (int mt = 0; mt < 2; ++mt) {
#pragma unroll
    for (int nt = 0; nt < 4; ++nt) {
      const int col  = bn + wn * 64 + nt * 16 + lo;
      const int rowb = bm + wm * 32 + mt * 16 + 8 * hi;
#pragma unroll
      for (int v = 0; v < 8; ++v)
        out[(size_t)(rowb + v) * H + col] = acc[mt][nt][v];
    }
  }
}

// ---------------------------------------------------------------------------
extern "C" void kernel_launch(void* const* d_in, const int* in_sizes, int n_in,
                              void* d_out, int out_size, void* d_ws, size_t ws_size,
                              hipStream_t stream) {
  const float* x      = (const float*)d_in[0];
  const float* weight = (const float*)d_in[1];
  const float* bias   = (const float*)d_in[2];
  const int*   dale   = (const int*)d_in[3];   // broadcast rows; row 0 used
  const int*   sparse = (const int*)d_in[4];
  float*       out    = (float*)d_out;
  float*       wp     = (float*)d_ws;          // H*H*4 = 64 MB scratch

  (void)in_sizes; (void)n_in; (void)out_size; (void)ws_size;

  // Phase 1: materialize masked/clipped weight
  const int n_vec_blocks = (H * H) / (256 * 4);
  mask_weights_kernel<<<n_vec_blocks, 256, 0, stream>>>(weight, sparse, dale, wp);

  // Phase 2: GEMM with bias
  dim3 grid(NB / BM, H / BN);
  dale_gemm_kernel<<<grid, 256, 0, stream>>>(x, wp, bias, out);
}